// GumbelTreeLSTMEncoder_5858335392100
// MI455X (gfx1250) — compile-verified
//
#include <hip/hip_runtime.h>
#include <hip/hip_bf16.h>
#include <math.h>

// ---------------------------------------------------------------------------
// Gumbel Tree-LSTM encoder for MI455X (gfx1250), FP32 WMMA (V_WMMA_F32_16X16X4_F32)
// B=64, L=24, D=512, H=512.  Comp GEMM has the LSTM gate epilogue fused.
// ---------------------------------------------------------------------------

typedef __attribute__((ext_vector_type(2))) float v2f;
typedef __attribute__((ext_vector_type(8))) float v8f;

#define BATCH 64
#define SEQL  24
#define HDIM  512

__device__ __forceinline__ float sigm(float x) { return 1.0f / (1.0f + __expf(-x)); }
__device__ __forceinline__ float tanh_fast(float x) { return 2.0f / (1.0f + __expf(-2.0f * x)) - 1.0f; }

// ---------------------------------------------------------------------------
// FP32 WMMA GEMM, W given row-major (N,K) (native weight layout).
//   MODE 0 (word): X = inp (R=1536,K=512). grid.y tiles 1024 cols by 256.
//                  cols<512 -> h_out, else c_out.  Plain bias epilogue.
//   MODE 1 (comp): X row r=(b,p) = [h[b,p,:] | h[b,p+1,:]] (K=1024).
//                  grid.y tiles the 512 intra-gate cols by 64; each wave
//                  accumulates ALL 5 gates for its 16 cols (5x v8f accs),
//                  then applies the LSTM nonlinearity inline and writes
//                  new_h/new_c directly.
// Workgroup: 256 threads = 8 waves as 2(M)x4(N); M tile 32; K staged 32/chunk.
// B staged in [n][k] layout: fragment = contiguous ds b64, bank-conflict-free.
// ---------------------------------------------------------------------------
template <int MODE>
__global__ __launch_bounds__(256) void gemm_wmma(const float* __restrict__ X,
                                                 const float* __restrict__ hbuf,
                                                 const float* __restrict__ cbuf,
                                                 const float* __restrict__ W,   // (N,K) row-major
                                                 const float* __restrict__ bias,
                                                 float* __restrict__ out0,
                                                 float* __restrict__ out1,
                                                 int K, int n) {
  __shared__ float A_lds[32 * 36];    // [row][k], padded stride 36
  __shared__ float B_lds[320 * 36];   // [n][k],   padded stride 36 (mode0 uses 256 rows)

  const int tid  = threadIdx.x;
  const unsigned lane = tid & 31u;
  const int wave = tid >> 5;          // 0..7
  const int wm   = wave >> 2;         // 0..1  (M half)
  const int wn   = wave & 3;          // 0..3  (N quarter)
  const int l16  = (int)(lane & 15u);
  const int lhi  = (int)(lane >> 4);  // 0 or 1

  const int r0 = blockIdx.x * 32;
  const int NACC = (MODE == 0) ? 4 : 5;

  v8f acc[5] = {};

  for (int k0 = 0; k0 < K; k0 += 32) {
    // ---- stage A chunk (32 rows x 32 k) : one float4 per thread ----
    {
      int ar  = tid >> 3;          // 0..31
      int ac4 = tid & 7;           // 0..7
      int rg  = r0 + ar;
      int kg  = k0 + ac4 * 4;
      float4 val;
      if (MODE == 0) {
        val = *(const float4*)&X[(size_t)rg * K + kg];
      } else {
        int b = rg / n;
        int p = rg - b * n;
        const float* src = (kg < HDIM)
            ? &hbuf[((size_t)(b * SEQL + p) * HDIM) + kg]
            : &hbuf[((size_t)(b * SEQL + p + 1) * HDIM) + (kg - HDIM)];
        val = *(const float4*)src;
      }
      *(float4*)&A_lds[ar * 36 + ac4 * 4] = val;
    }
    // ---- stage B chunk from native (N,K) weights ----
    if (MODE == 0) {
      // 256 cols starting at blockIdx.y*256 : 2048 float4
      int nb0 = blockIdx.y * 256;
#pragma unroll
      for (int i = 0; i < 8; ++i) {
        int f    = i * 256 + tid;
        int nrow = f >> 3;           // 0..255
        int kc4  = f & 7;
        float4 val = *(const float4*)&W[(size_t)(nb0 + nrow) * K + k0 + kc4 * 4];
        *(float4*)&B_lds[nrow * 36 + kc4 * 4] = val;
      }
    } else {
      // 5 gates x 64 intra-gate cols starting at blockIdx.y*64 : 2560 float4
      int nb0 = blockIdx.y * 64;
#pragma unroll
      for (int i = 0; i < 10; ++i) {
        int f    = i * 256 + tid;
        int nrow = f >> 3;           // 0..319
        int kc4  = f & 7;
        int g    = nrow >> 6;        // gate 0..4
        int nl   = nrow & 63;        // intra-gate col 0..63
        float4 val = *(const float4*)&W[(size_t)(g * HDIM + nb0 + nl) * K + k0 + kc4 * 4];
        *(float4*)&B_lds[nrow * 36 + kc4 * 4] = val;
      }
    }
    __syncthreads();

    // ---- 8 WMMA k-steps of K=4 ----
#pragma unroll
    for (int kk = 0; kk < 32; kk += 4) {
      // A frag 16x4: lanes 0-15 -> K=kk,kk+1 ; lanes 16-31 -> K=kk+2,kk+3
      v2f a = *(const v2f*)&A_lds[(wm * 16 + l16) * 36 + kk + 2 * lhi];
#pragma unroll
      for (int g = 0; g < NACC; ++g) {
        int nrow = (MODE == 0) ? (wn * 64 + g * 16 + l16)
                               : (g * 64 + wn * 16 + l16);
        v2f bf = *(const v2f*)&B_lds[nrow * 36 + kk + 2 * lhi];
        acc[g] = __builtin_amdgcn_wmma_f32_16x16x4_f32(
            false, a, false, bf, (short)0, acc[g], false, false);
      }
    }
    __syncthreads();
  }

  // ---- epilogue ----
  if (MODE == 0) {
    const int colbase = blockIdx.y * 256 + wn * 64;
    const int rowg0   = r0 + wm * 16 + 8 * lhi;
#pragma unroll
    for (int j = 0; j < 4; ++j) {
      int col = colbase + j * 16 + l16;
      float bv = bias[col];
#pragma unroll
      for (int vr = 0; vr < 8; ++vr) {
        int rg = rowg0 + vr;
        float val = acc[j][vr] + bv;
        if (col < HDIM) out0[(size_t)rg * HDIM + col] = val;
        else            out1[(size_t)rg * HDIM + (col - HDIM)] = val;
      }
    }
  } else {
    // fused LSTM gate epilogue: acc[0..4] = gi, gfl, gfr, gu, go (pre-bias)
    const int colg = blockIdx.y * 64 + wn * 16 + l16;
    float bz[5];
#pragma unroll
    for (int g = 0; g < 5; ++g) bz[g] = bias[g * HDIM + colg];
    int rg = r0 + wm * 16 + 8 * lhi;
    int b  = rg / n;
    int p  = rg - b * n;
#pragma unroll
    for (int vr = 0; vr < 8; ++vr) {
      float gi  = acc[0][vr] + bz[0];
      float gfl = acc[1][vr] + bz[1];
      float gfr = acc[2][vr] + bz[2];
      float gu  = acc[3][vr] + bz[3];
      float go  = acc[4][vr] + bz[4];
      size_t cbase = (size_t)(b * SEQL + p) * HDIM + colg;
      float cl = cbuf[cbase];
      float cr = cbuf[cbase + HDIM];
      float nc = cl * sigm(gfl + 1.0f) + cr * sigm(gfr + 1.0f) + tanh_fast(gu) * sigm(gi);
      float nh = sigm(go) * tanh_fast(nc);
      size_t oi = (size_t)rg * HDIM + colg;
      out0[oi] = nh;   // new_h
      out1[oi] = nc;   // new_c
      ++rg; ++p;
      if (p == n) { p = 0; ++b; }
    }
  }
}

// ---------------------------------------------------------------------------
// Per-batch: scores -> masked argmax -> sel/left/right mix -> done mask.
// One 256-thread block per batch element.
// ---------------------------------------------------------------------------
__global__ __launch_bounds__(256) void mix_k(const float* __restrict__ newh,
                                             const float* __restrict__ newc,
                                             const float* __restrict__ hcur,
                                             const float* __restrict__ ccur,
                                             const float* __restrict__ q,
                                             const int* __restrict__ length,
                                             float* __restrict__ hnxt,
                                             float* __restrict__ cnxt,
                                             int n, int iter) {
  __shared__ float sc[32];
  __shared__ int ksel_s;
  const int b = blockIdx.x;
  const int tid = threadIdx.x;
  const unsigned lane = tid & 31u;
  const int wave = tid >> 5;

  // scores[p] = dot(new_h[b,p,:], q), one wave per position
  for (int p = wave; p < n; p += 8) {
    const float* row = newh + (size_t)(b * n + p) * HDIM;
    float s = 0.0f;
    for (int j = (int)lane; j < HDIM; j += 32) s += row[j] * q[j];
#pragma unroll
    for (int off = 16; off > 0; off >>= 1) s += __shfl_xor(s, off, 32);
    if (lane == 0) sc[p] = s;
  }
  __syncthreads();

  if (tid == 0) {
    int len = length[b];
    int best = 0;
    float bv = -INFINITY;
    for (int p = 0; p < n; ++p) {
      float s = ((iter + 1 + p) < len) ? sc[p] : -1.0e9f;
      if (s > bv) { bv = s; best = p; }  // strict '>' => first max, like jnp.argmax
    }
    ksel_s = best;
  }
  __syncthreads();

  const int ksel = ksel_s;
  const bool done = (iter + 1) < length[b];

  for (int idx = tid; idx < n * HDIM; idx += 256) {
    int p = idx >> 9;
    int j = idx & (HDIM - 1);
    size_t cur = (size_t)(b * SEQL + p) * HDIM + j;
    float hm, cm;
    if (p < ksel)      { hm = hcur[cur];        cm = ccur[cur]; }
    else if (p > ksel) { hm = hcur[cur + HDIM]; cm = ccur[cur + HDIM]; }
    else {
      size_t ni = (size_t)(b * n + p) * HDIM + j;
      hm = newh[ni]; cm = newc[ni];
    }
    hnxt[cur] = done ? hm : hcur[cur];
    cnxt[cur] = done ? cm : ccur[cur];
  }
}

// ---------------------------------------------------------------------------
__global__ __launch_bounds__(256) void final_copy_k(const float* __restrict__ hfin,
                                                    float* __restrict__ out) {
  int idx = blockIdx.x * 256 + threadIdx.x;  // 64*512 total
  int b = idx >> 9;
  int j = idx & (HDIM - 1);
  out[idx] = hfin[(size_t)b * SEQL * HDIM + j];
}

// ---------------------------------------------------------------------------
extern "C" void kernel_launch(void* const* d_in, const int* in_sizes, int n_in,
                              void* d_out, int out_size, void* d_ws, size_t ws_size,
                              hipStream_t stream) {
  const float* inp    = (const float*)d_in[0];  // (64,24,512)
  const int*   length = (const int*)  d_in[1];  // (64,)
  const float* word_W = (const float*)d_in[2];  // (1024,512)   (N,K)
  const float* word_b = (const float*)d_in[3];  // (1024,)
  const float* comp_W = (const float*)d_in[4];  // (2560,1024)  (N,K)
  const float* comp_b = (const float*)d_in[5];  // (2560,)
  const float* comp_q = (const float*)d_in[6];  // (512,)

  float* ws = (float*)d_ws;
  size_t off = 0;
  float* hb[2];
  float* cb[2];
  hb[0] = ws + off; off += (size_t)BATCH * SEQL * HDIM;
  cb[0] = ws + off; off += (size_t)BATCH * SEQL * HDIM;
  hb[1] = ws + off; off += (size_t)BATCH * SEQL * HDIM;
  cb[1] = ws + off; off += (size_t)BATCH * SEQL * HDIM;
  float* newh = ws + off; off += (size_t)(BATCH * 23) * HDIM;
  float* newc = ws + off; off += (size_t)(BATCH * 23) * HDIM;

  // word projection: (1536,512)@(512,1024) -> split h/c
  gemm_wmma<0><<<dim3((BATCH * SEQL) / 32, 1024 / 256), 256, 0, stream>>>(
      inp, nullptr, nullptr, word_W, word_b, hb[0], cb[0], 512, 0);

  int cur = 0;
  for (int i = 0; i < SEQL - 1; ++i) {
    int n = (SEQL - 1) - i;  // 23..1
    int R = BATCH * n;
    int nxt = cur ^ 1;
    // comp GEMM with fused LSTM gates -> new_h, new_c
    gemm_wmma<1><<<dim3(R / 32, HDIM / 64), 256, 0, stream>>>(
        nullptr, hb[cur], cb[cur], comp_W, comp_b, newh, newc, 1024, n);
    mix_k<<<BATCH, 256, 0, stream>>>(newh, newc, hb[cur], cb[cur], comp_q, length,
                                     hb[nxt], cb[nxt], n, i);
    cur = nxt;
  }

  final_copy_k<<<(BATCH * HDIM) / 256, 256, 0, stream>>>(hb[cur], (float*)d_out);
  (void)in_sizes; (void)n_in; (void)out_size; (void)ws_size;
}